// SelectiveSSM_76888504533255
// MI455X (gfx1250) — compile-verified
//
#include <hip/hip_runtime.h>
#include <hip/hip_bf16.h>
#include <math.h>

// ---------------------------------------------------------------------------
// Mamba selective-SSM forward for MI455X (gfx1250), fp32 end-to-end.
//  * GEMMs: V_WMMA_F32_16X16X4_F32, 128x128 block tile, 32x64 wave tile,
//    double-buffered LDS panels fed by GLOBAL_LOAD_ASYNC_TO_LDS_B128
//    (ASYNCcnt) when available.
//  * Scan: 8-chunk linear-recurrence decomposition; chunk propagator is
//    exp(A * sum(delta)) by linearity, so the scan is 8x parallel.
// ---------------------------------------------------------------------------

typedef float v2f __attribute__((ext_vector_type(2)));
typedef float v8f __attribute__((ext_vector_type(8)));
typedef int   v4i_vs __attribute__((vector_size(16)));   // matches builtin param

#define DMODEL 1024
#define DSTATE 16
#define DCONV  4
#define DINNER 2048
#define DTRANK 64
#define BATCH  4
#define SEQ    1024
#define MROWS  (BATCH * SEQ)     // 4096 flattened (b,l) rows
#define NCHUNK 8
#define CHUNKL (SEQ / NCHUNK)    // 128

#if defined(__has_builtin)
#  if __has_builtin(__builtin_amdgcn_global_load_async_to_lds_b128) && \
      __has_builtin(__builtin_amdgcn_s_wait_asynccnt)
#    define HAVE_ASYNC_LDS 1
#  endif
#endif
#ifndef HAVE_ASYNC_LDS
#  define HAVE_ASYNC_LDS 0
#endif

__device__ __forceinline__ float sigmoid_f(float v) {
    return 1.0f / (1.0f + __expf(-v));
}

#if HAVE_ASYNC_LDS
// 16B global -> LDS async copy (per-lane, EXEC-predicated, ASYNCcnt-tracked)
__device__ __forceinline__ void async_cp16(const float* g, float* l) {
    __builtin_amdgcn_global_load_async_to_lds_b128(
        (__attribute__((address_space(1))) v4i_vs*)const_cast<float*>(g),
        (__attribute__((address_space(3))) v4i_vs*)l, 0, 0);
}
#endif

// ---------------------------------------------------------------------------
// fp32 WMMA GEMM: C[M,N] = A[M,K] @ B[K,N], row-major, strides given.
// Block = 256 threads = 8 wave32 waves. Block tile 128(M) x 128(N), K panel 32.
// Wave tile 32x64: acc[2][4] of v8f; every B fragment feeds 2 WMMAs.
// EPI==1: C = softplus(C + bias[col])   (delta branch)
// ---------------------------------------------------------------------------
template <int EPI>
__global__ __launch_bounds__(256)
void gemm_wmma_f32(const float* __restrict__ A, const float* __restrict__ B,
                   float* __restrict__ C, const float* __restrict__ bias,
                   int M, int N, int K, int lda, int ldb, int ldc)
{
    // stride 36 / 132: multiples of 4 (16B-aligned b128 stores), and 36 is
    // coprime-enough with 64 banks that 16 row-strided lanes hit 16 banks.
    __shared__ float As[2][128][36];
    __shared__ float Bs[2][32][132];

    const int tid  = threadIdx.x;
    const int lane = tid & 31;
    const int wave = tid >> 5;           // 0..7
    const int l15  = lane & 15;
    const int hi   = lane >> 4;          // 0: lanes 0-15, 1: lanes 16-31
    const int wr   = (wave >> 1) * 32;   // wave row offset in block tile
    const int wc   = (wave & 1) * 64;    // wave col offset in block tile
    const int rowBase = blockIdx.y * 128;
    const int colBase = blockIdx.x * 128;

    v8f acc[2][4];
#pragma unroll
    for (int s = 0; s < 2; ++s)
#pragma unroll
        for (int t = 0; t < 4; ++t)
#pragma unroll
            for (int r = 0; r < 8; ++r) acc[s][t][r] = 0.0f;

    // ---- panel loader: A 128x32, B 32x128 into buffer `buf` ----
    auto load_panels = [&](int k0, int buf) {
#if HAVE_ASYNC_LDS
#pragma unroll
        for (int i = 0; i < 4; ++i) {
            int e = i * 256 + tid;               // 1024 float4 slots for A
            int r = e >> 3;                      // 0..127
            int c = (e & 7) * 4;                 // 0..28
            async_cp16(&A[(size_t)(rowBase + r) * lda + k0 + c], &As[buf][r][c]);
        }
#pragma unroll
        for (int i = 0; i < 4; ++i) {
            int e = i * 256 + tid;               // 1024 float4 slots for B
            int r = e >> 5;                      // 0..31
            int c = (e & 31) * 4;                // 0..124
            int gcol = colBase + c;
            if (gcol < N)                        // dead cols: stale LDS, never stored
                async_cp16(&B[(size_t)(k0 + r) * ldb + gcol], &Bs[buf][r][c]);
        }
#else
#pragma unroll
        for (int i = 0; i < 4; ++i) {
            int e = i * 256 + tid;
            int r = e >> 3;
            int c = (e & 7) * 4;
            const float4 v =
                *(const float4*)&A[(size_t)(rowBase + r) * lda + k0 + c];
            As[buf][r][c + 0] = v.x; As[buf][r][c + 1] = v.y;
            As[buf][r][c + 2] = v.z; As[buf][r][c + 3] = v.w;
        }
#pragma unroll
        for (int i = 0; i < 4; ++i) {
            int e = i * 256 + tid;
            int r = e >> 5;
            int c = (e & 31) * 4;
            int gcol = colBase + c;
            float4 v = make_float4(0.f, 0.f, 0.f, 0.f);
            if (gcol < N)
                v = *(const float4*)&B[(size_t)(k0 + r) * ldb + gcol];
            Bs[buf][r][c + 0] = v.x; Bs[buf][r][c + 1] = v.y;
            Bs[buf][r][c + 2] = v.z; Bs[buf][r][c + 3] = v.w;
        }
#endif
    };
    auto wait_transfers = [&]() {
#if HAVE_ASYNC_LDS
        __builtin_amdgcn_s_wait_asynccnt(0);
#endif
    };

    const int nk = K >> 5;                        // K / 32 panels
    load_panels(0, 0);
    wait_transfers();
    __syncthreads();

    for (int ik = 0; ik < nk; ++ik) {
        const int cur = ik & 1;
        if (ik + 1 < nk)
            load_panels((ik + 1) << 5, cur ^ 1);  // overlap with compute below

        // ---- 8 WMMA K-steps of 4 over the 32-wide panel ----
#pragma unroll
        for (int kk = 0; kk < 8; ++kk) {
            const int ka = kk * 4 + hi * 2;       // K pair for this half-wave
            v2f a0, a1;
            a0.x = As[cur][wr + l15][ka];
            a0.y = As[cur][wr + l15][ka + 1];
            a1.x = As[cur][wr + 16 + l15][ka];
            a1.y = As[cur][wr + 16 + l15][ka + 1];
#pragma unroll
            for (int t = 0; t < 4; ++t) {
                v2f bf;
                bf.x = Bs[cur][ka][wc + t * 16 + l15];
                bf.y = Bs[cur][ka + 1][wc + t * 16 + l15];
                acc[0][t] = __builtin_amdgcn_wmma_f32_16x16x4_f32(
                    false, a0, false, bf, (short)0, acc[0][t], false, false);
                acc[1][t] = __builtin_amdgcn_wmma_f32_16x16x4_f32(
                    false, a1, false, bf, (short)0, acc[1][t], false, false);
            }
        }

        if (ik + 1 < nk) {
            wait_transfers();
            __syncthreads();
        }
    }

    // ---- epilogue / store: VGPR r -> M=r (lanes 0-15) / M=r+8 (lanes 16-31)
#pragma unroll
    for (int t = 0; t < 4; ++t) {
        const int col = colBase + wc + t * 16 + l15;
        if (col < N) {
            const float bv = (EPI == 1) ? bias[col] : 0.0f;
#pragma unroll
            for (int s = 0; s < 2; ++s) {
#pragma unroll
                for (int r = 0; r < 8; ++r) {
                    const int row = rowBase + wr + s * 16 + hi * 8 + r;
                    float v = acc[s][t][r];
                    if (EPI == 1) {
                        v += bv;
                        v = (v > 20.0f) ? v : log1pf(__expf(v));   // softplus
                    }
                    C[(size_t)row * ldc + col] = v;
                }
            }
        }
    }
}

// ---------------------------------------------------------------------------
// Causal depthwise conv1d (width 4) + bias + SiLU over the xi half of xz.
// ---------------------------------------------------------------------------
__global__ __launch_bounds__(256)
void conv_silu(const float* __restrict__ xz, const float* __restrict__ cw,
               const float* __restrict__ cb, float* __restrict__ xic)
{
    const int tid = threadIdx.x;
    const int b   = blockIdx.z;
    const int l0  = blockIdx.y * 16;
    const int d   = blockIdx.x * 256 + tid;

    const float w0 = cw[d * 4 + 0], w1 = cw[d * 4 + 1];
    const float w2 = cw[d * 4 + 2], w3 = cw[d * 4 + 3];
    const float bb = cb[d];

    const size_t rs   = 2 * DINNER;                  // xz row stride
    const size_t base = (size_t)b * SEQ * rs + d;

    float x0 = (l0 >= 3) ? xz[base + (size_t)(l0 - 3) * rs] : 0.0f;
    float x1 = (l0 >= 2) ? xz[base + (size_t)(l0 - 2) * rs] : 0.0f;
    float x2 = (l0 >= 1) ? xz[base + (size_t)(l0 - 1) * rs] : 0.0f;

#pragma unroll
    for (int i = 0; i < 16; ++i) {
        const int l = l0 + i;
        const float xl = xz[base + (size_t)l * rs];
        float v = w0 * x0 + w1 * x1 + w2 * x2 + w3 * xl + bb;
        v = v * sigmoid_f(v);                        // SiLU
        xic[((size_t)b * SEQ + l) * DINNER + d] = v;
        x0 = x1; x1 = x2; x2 = xl;
    }
}

// ---------------------------------------------------------------------------
// Scan pass 1: per (b, chunk, d) run recurrence from h=0 over 128 steps,
// record local end-state and sum of delta. B_t vectors staged in LDS.
// ---------------------------------------------------------------------------
__global__ __launch_bounds__(256)
void ssm_scan_pass1(const float* __restrict__ delta, const float* __restrict__ xic,
                    const float* __restrict__ xdbl, const float* __restrict__ Alog,
                    float* __restrict__ hlocal, float* __restrict__ sumd)
{
    __shared__ float sB[CHUNKL][DSTATE];
    const int tid = threadIdx.x;
    const int b = blockIdx.z, c = blockIdx.y;
    const int d = blockIdx.x * 256 + tid;
    const int l0 = c * CHUNKL;

    for (int e = tid; e < CHUNKL * DSTATE; e += 256) {
        int li = e >> 4, n = e & 15;
        sB[li][n] = xdbl[((size_t)b * SEQ + l0 + li) * 96 + DTRANK + n];
    }
    __syncthreads();

    float Av[DSTATE];
#pragma unroll
    for (int n = 0; n < DSTATE; ++n) Av[n] = -__expf(Alog[d * DSTATE + n]);

    float h[DSTATE];
#pragma unroll
    for (int n = 0; n < DSTATE; ++n) h[n] = 0.0f;
    float sd = 0.0f;

    for (int li = 0; li < CHUNKL; ++li) {
        const size_t row = ((size_t)b * SEQ + l0 + li) * DINNER + d;
        const float dt = delta[row];
        const float xv = xic[row];
        sd += dt;
        const float du = dt * xv;
#pragma unroll
        for (int n = 0; n < DSTATE; ++n)
            h[n] = __expf(dt * Av[n]) * h[n] + du * sB[li][n];
    }

    const size_t base = (((size_t)b * NCHUNK + c) * DINNER + d) * DSTATE;
#pragma unroll
    for (int n = 0; n < DSTATE; ++n) hlocal[base + n] = h[n];
    sumd[((size_t)b * NCHUNK + c) * DINNER + d] = sd;
}

// ---------------------------------------------------------------------------
// Chunk combine: h_init[c] = P_{c-1} h_init[c-1] + hlocal[c-1],
// with P_c[n] = exp(A[n] * sum(delta over chunk c))  (linearity of the SSM).
// ---------------------------------------------------------------------------
__global__ __launch_bounds__(256)
void ssm_combine(const float* __restrict__ hlocal, const float* __restrict__ sumd,
                 const float* __restrict__ Alog, float* __restrict__ hinit)
{
    const int tid = threadIdx.x;
    const int b = blockIdx.y;
    const int d = blockIdx.x * 256 + tid;

    float Av[DSTATE];
#pragma unroll
    for (int n = 0; n < DSTATE; ++n) Av[n] = -__expf(Alog[d * DSTATE + n]);

    float h[DSTATE];
#pragma unroll
    for (int n = 0; n < DSTATE; ++n) h[n] = 0.0f;

    for (int c = 0; c < NCHUNK; ++c) {
        const size_t base = (((size_t)b * NCHUNK + c) * DINNER + d) * DSTATE;
#pragma unroll
        for (int n = 0; n < DSTATE; ++n) hinit[base + n] = h[n];
        const float sd = sumd[((size_t)b * NCHUNK + c) * DINNER + d];
#pragma unroll
        for (int n = 0; n < DSTATE; ++n)
            h[n] = __expf(Av[n] * sd) * h[n] + hlocal[base + n];
    }
}

// ---------------------------------------------------------------------------
// Scan pass 2: replay recurrence from h_init, emit y = h.C + Dp*xi, gate with
// silu(z). B_t and C_t staged in LDS.
// ---------------------------------------------------------------------------
__global__ __launch_bounds__(256)
void ssm_scan_pass2(const float* __restrict__ delta, const float* __restrict__ xic,
                    const float* __restrict__ xdbl, const float* __restrict__ Alog,
                    const float* __restrict__ hinit, const float* __restrict__ Dp,
                    const float* __restrict__ xz, float* __restrict__ yg)
{
    __shared__ float sB[CHUNKL][DSTATE];
    __shared__ float sC[CHUNKL][DSTATE];
    const int tid = threadIdx.x;
    const int b = blockIdx.z, c = blockIdx.y;
    const int d = blockIdx.x * 256 + tid;
    const int l0 = c * CHUNKL;

    for (int e = tid; e < CHUNKL * DSTATE; e += 256) {
        int li = e >> 4, n = e & 15;
        const size_t rowb = ((size_t)b * SEQ + l0 + li) * 96 + DTRANK;
        sB[li][n] = xdbl[rowb + n];
        sC[li][n] = xdbl[rowb + DSTATE + n];
    }
    __syncthreads();

    float Av[DSTATE];
#pragma unroll
    for (int n = 0; n < DSTATE; ++n) Av[n] = -__expf(Alog[d * DSTATE + n]);

    float h[DSTATE];
    const size_t hb = (((size_t)b * NCHUNK + c) * DINNER + d) * DSTATE;
#pragma unroll
    for (int n = 0; n < DSTATE; ++n) h[n] = hinit[hb + n];

    const float Dpd = Dp[d];

    for (int li = 0; li < CHUNKL; ++li) {
        const int l = l0 + li;
        const size_t row = ((size_t)b * SEQ + l) * DINNER + d;
        const float dt = delta[row];
        const float xv = xic[row];
        const float du = dt * xv;
        float yt = 0.0f;
#pragma unroll
        for (int n = 0; n < DSTATE; ++n) {
            h[n] = __expf(dt * Av[n]) * h[n] + du * sB[li][n];
            yt += h[n] * sC[li][n];
        }
        float y = yt + Dpd * xv;
        const float zv = xz[((size_t)b * SEQ + l) * (2 * DINNER) + DINNER + d];
        yg[row] = y * (zv * sigmoid_f(zv));
    }
}

// ---------------------------------------------------------------------------
// Host-side orchestration (all on `stream`, graph-capture safe).
// Inputs (dict order): x, W_in, conv_w, conv_b, W_x, W_dt, b_dt, A_log, Dp, W_out
// ---------------------------------------------------------------------------
extern "C" void kernel_launch(void* const* d_in, const int* in_sizes, int n_in,
                              void* d_out, int out_size, void* d_ws, size_t ws_size,
                              hipStream_t stream)
{
    const float* x    = (const float*)d_in[0];
    const float* Win  = (const float*)d_in[1];
    const float* cw   = (const float*)d_in[2];
    const float* cb   = (const float*)d_in[3];
    const float* Wx   = (const float*)d_in[4];
    const float* Wdt  = (const float*)d_in[5];
    const float* bdt  = (const float*)d_in[6];
    const float* Alog = (const float*)d_in[7];
    const float* Dp   = (const float*)d_in[8];
    const float* Wout = (const float*)d_in[9];
    float* out = (float*)d_out;

    char* p = (char*)d_ws;
    auto carve = [&](size_t bytes) -> float* {
        float* r = (float*)p;
        p += (bytes + 255) & ~(size_t)255;
        return r;
    };
    float* xz    = carve((size_t)MROWS * 2 * DINNER * sizeof(float));   // 64 MB
    float* xic   = carve((size_t)MROWS * DINNER * sizeof(float));       // 32 MB
    float* xdbl  = carve((size_t)MROWS * 96 * sizeof(float));           // 1.5 MB
    float* dlt   = carve((size_t)MROWS * DINNER * sizeof(float));       // 32 MB
    float* yg    = carve((size_t)MROWS * DINNER * sizeof(float));       // 32 MB
    float* hloc  = carve((size_t)BATCH * NCHUNK * DINNER * DSTATE * sizeof(float));
    float* hini  = carve((size_t)BATCH * NCHUNK * DINNER * DSTATE * sizeof(float));
    float* sumd  = carve((size_t)BATCH * NCHUNK * DINNER * sizeof(float));

    const dim3 gblk(256);

    // 1) xz = x @ W_in                           (4096 x 1024 x 4096)
    gemm_wmma_f32<0><<<dim3(2 * DINNER / 128, MROWS / 128), gblk, 0, stream>>>(
        x, Win, xz, nullptr, MROWS, 2 * DINNER, DMODEL, DMODEL, 2 * DINNER, 2 * DINNER);

    // 2) causal depthwise conv + SiLU on xi half
    conv_silu<<<dim3(DINNER / 256, SEQ / 16, BATCH), 256, 0, stream>>>(xz, cw, cb, xic);

    // 3) x_dbl = xi_c @ W_x                      (4096 x 2048 x 96)
    gemm_wmma_f32<0><<<dim3(1, MROWS / 128), gblk, 0, stream>>>(
        xic, Wx, xdbl, nullptr, MROWS, 96, DINNER, DINNER, 96, 96);

    // 4) delta = softplus(x_dbl[:, :64] @ W_dt + b_dt)   (4096 x 64 x 2048)
    gemm_wmma_f32<1><<<dim3(DINNER / 128, MROWS / 128), gblk, 0, stream>>>(
        xdbl, Wdt, dlt, bdt, MROWS, DINNER, DTRANK, 96, DINNER, DINNER);

    // 5-7) chunked selective scan
    ssm_scan_pass1<<<dim3(DINNER / 256, NCHUNK, BATCH), 256, 0, stream>>>(
        dlt, xic, xdbl, Alog, hloc, sumd);
    ssm_combine<<<dim3(DINNER / 256, BATCH), 256, 0, stream>>>(hloc, sumd, Alog, hini);
    ssm_scan_pass2<<<dim3(DINNER / 256, NCHUNK, BATCH), 256, 0, stream>>>(
        dlt, xic, xdbl, Alog, hini, Dp, xz, yg);

    // 8) out = yg @ W_out                        (4096 x 2048 x 1024)
    gemm_wmma_f32<0><<<dim3(DMODEL / 128, MROWS / 128), gblk, 0, stream>>>(
        yg, Wout, out, nullptr, MROWS, DMODEL, DINNER, DINNER, DMODEL, DMODEL);
}